// RAP_57999238365252
// MI455X (gfx1250) — compile-verified
//
#include <hip/hip_runtime.h>
#include <stdint.h>

#define D_DIM 1024
#define N_DIM 65536
#define Q_DIM 2048
#define TILE_N 64
#define ROW_STRIDE 65                 // 64 data dwords + 1 pad dword (bank-conflict-free gathers)
#define NBLOCKS (N_DIM / TILE_N)      // 1024

typedef float v2f __attribute__((ext_vector_type(2)));
typedef float v8f __attribute__((ext_vector_type(8)));
typedef unsigned int uv4 __attribute__((ext_vector_type(4)));
typedef int iv4 __attribute__((ext_vector_type(4)));
typedef int iv8 __attribute__((ext_vector_type(8)));

// Issue one TDM 2D tile load: rows [row0, row0+nrows) of weight columns [n0, n0+64) -> LDS,
// with 1-dword pad after every 64 dwords so LDS rows have stride ROW_STRIDE.
__device__ __forceinline__ void tdm_load_half(const float* weight, int n0, int row0, int nrows,
                                              uint32_t lds_base_bytes) {
  uint64_t gaddr = (uint64_t)(uintptr_t)(weight + (size_t)row0 * N_DIM + n0);
  uint32_t laddr = lds_base_bytes + (uint32_t)row0 * ROW_STRIDE * 4u;
  uv4 g0;
  g0[0] = 1u;                                                   // count=1 (user descriptor)
  g0[1] = laddr;                                                // lds_addr (bytes)
  g0[2] = (uint32_t)gaddr;                                      // global_addr[31:0]
  g0[3] = (uint32_t)((gaddr >> 32) & 0x01FFFFFFu) | (2u << 30); // global_addr[56:32] | type=2
  iv8 g1;
  g1[0] = (int)((2u << 16)      // data_size = 4 bytes
              | (1u << 20)      // pad_enable
              | (5u << 22)      // pad_interval: 2<<5 = 64 dwords
              | (0u << 25));    // pad_amount: 1 dword
  g1[1] = (int)((N_DIM & 0xFFFFu) << 16);                         // tensor_dim0 lo16 (65536 -> 0)
  g1[2] = (int)((unsigned)(N_DIM >> 16) | ((unsigned)(D_DIM & 0xFFFF) << 16)); // dim0 hi16 | dim1 lo16
  g1[3] = (int)((unsigned)(D_DIM >> 16) | ((unsigned)TILE_N << 16));           // dim1 hi16 | tile_dim0
  g1[4] = nrows;                                                 // tile_dim1 (tile_dim2 = 0)
  g1[5] = N_DIM;                                                 // tensor_dim0_stride lo32
  g1[6] = 0;
  g1[7] = 0;
  iv4 gz = {0, 0, 0, 0};
#if __clang_major__ >= 23
  iv8 gz8 = {0, 0, 0, 0, 0, 0, 0, 0};
  __builtin_amdgcn_tensor_load_to_lds(g0, g1, gz, gz, gz8, 0);
#else
  __builtin_amdgcn_tensor_load_to_lds(g0, g1, gz, gz, 0);
#endif
}

__global__ __launch_bounds__(256) void RAP_probs_kernel(const float* __restrict__ weight,
                                                        const int* __restrict__ q_idxs,
                                                        float* __restrict__ partial) {
  __shared__ float probs[D_DIM * ROW_STRIDE];   // 266,240 B
  __shared__ int   qidx[Q_DIM * 3];             //  24,576 B   (total 290,816 B <= 320 KB)

  const int tid = threadIdx.x;
  const int blk = blockIdx.x;
  const int n0  = blk * TILE_N;

  // ---------- Phase 0: TDM async tile load (two halves for overlap) + stage q_idxs ----------
  if (tid < 32) {  // wave-uniform: only wave 0 issues the DMA
    uint32_t lds_base = (uint32_t)(uint64_t)(uintptr_t)(void*)&probs[0];
    tdm_load_half(weight, n0, 0,   512, lds_base);
    tdm_load_half(weight, n0, 512, 512, lds_base);
    __builtin_amdgcn_s_wait_tensorcnt(1);   // first half landed
  }
  for (int i = tid; i < Q_DIM * 3; i += 256) qidx[i] = q_idxs[i];
  __syncthreads();

  // ---------- Phase 1: in-place grouped softmax (64 groups of 16 bins per column) ----------
  // thread -> column nl = tid&63, quarter = tid>>6 handles groups a = g*4 + quarter.
  const int nl = tid & (TILE_N - 1);
  const int quarter = tid >> 6;
  for (int half = 0; half < 2; ++half) {
    if (half == 1) {
      if (tid < 32) __builtin_amdgcn_s_wait_tensorcnt(0);  // second half landed
      __syncthreads();
    }
    #pragma unroll
    for (int g = half * 8; g < half * 8 + 8; ++g) {
      const int a = g * 4 + quarter;            // attr group 0..63
      const int base = (a * 16) * ROW_STRIDE + nl;
      float v[16];
      float m = -3.402823466e38f;
      #pragma unroll
      for (int b = 0; b < 16; ++b) { v[b] = probs[base + b * ROW_STRIDE]; m = fmaxf(m, v[b]); }
      float s = 0.0f;
      #pragma unroll
      for (int b = 0; b < 16; ++b) { v[b] = __expf(v[b] - m); s += v[b]; }
      const float inv = 1.0f / s;
      #pragma unroll
      for (int b = 0; b < 16; ++b) probs[base + b * ROW_STRIDE] = v[b] * inv;
    }
  }
  __syncthreads();

  // ---------- Phase 2: gather triple-products, reduce over the 64 columns via WMMA ----------
  // A (16x4 f32): lane layout per ISA: lanes 0-15 row=lane, K={0,1}; lanes 16-31 row=lane-16, K={2,3}.
  // B = ones(4x16)  =>  D[m, j] = sum_k A[m, k] + C[m, j]  (column-replicated row sums).
  const int wave = tid >> 5;
  const int lane = tid & 31;
  const int mrow = lane & 15;
  const int koff = (lane >> 4) << 1;            // 0 or 2
  float* pblock = partial + (size_t)blk * Q_DIM;
  const v2f ones = {1.0f, 1.0f};

  for (int t = wave; t < Q_DIM / 16; t += 8) {
    const int q0 = t * 16;
    const int qm = q0 + mrow;
    const int i0 = qidx[qm * 3 + 0];
    const int i1 = qidx[qm * 3 + 1];
    const int i2 = qidx[qm * 3 + 2];
    const float* r0 = &probs[i0 * ROW_STRIDE];
    const float* r1 = &probs[i1 * ROW_STRIDE];
    const float* r2 = &probs[i2 * ROW_STRIDE];
    v8f acc = {0.f, 0.f, 0.f, 0.f, 0.f, 0.f, 0.f, 0.f};
    #pragma unroll
    for (int c = 0; c < TILE_N / 4; ++c) {
      const int x0 = c * 4 + koff;
      v2f a;
      a[0] = r0[x0]     * r1[x0]     * r2[x0];
      a[1] = r0[x0 + 1] * r1[x0 + 1] * r2[x0 + 1];
      acc = __builtin_amdgcn_wmma_f32_16x16x4_f32(false, a, false, ones,
                                                  (short)0, acc, false, false);
    }
    // D layout (16x16 f32): VGPR r: lanes 0-15 -> M=r, lanes 16-31 -> M=r+8; all columns equal.
    if (lane == 0) {
      #pragma unroll
      for (int m = 0; m < 8; ++m) pblock[q0 + m] = acc[m];
    } else if (lane == 16) {
      #pragma unroll
      for (int m = 0; m < 8; ++m) pblock[q0 + 8 + m] = acc[m];
    }
  }
}

// Deterministic cross-block reduction: out[q] = (1/N) * sum_b partial[b][q]
__global__ __launch_bounds__(256) void RAP_reduce_kernel(const float* __restrict__ partial,
                                                         float* __restrict__ out) {
  const int q = blockIdx.x * blockDim.x + threadIdx.x;
  if (q < Q_DIM) {
    float s = 0.0f;
    for (int b = 0; b < NBLOCKS; ++b) s += partial[(size_t)b * Q_DIM + q];
    out[q] = s * (1.0f / (float)N_DIM);
  }
}

extern "C" void kernel_launch(void* const* d_in, const int* in_sizes, int n_in,
                              void* d_out, int out_size, void* d_ws, size_t ws_size,
                              hipStream_t stream) {
  const float* weight = (const float*)d_in[0];
  const int*   q_idxs = (const int*)d_in[1];
  float* partial = (float*)d_ws;   // needs NBLOCKS * Q_DIM * 4 = 8 MB of scratch
  float* out = (float*)d_out;

  RAP_probs_kernel<<<NBLOCKS, 256, 0, stream>>>(weight, q_idxs, partial);
  RAP_reduce_kernel<<<(Q_DIM + 255) / 256, 256, 0, stream>>>(partial, out);
}